// OmniSVGModel_32195074850876
// MI455X (gfx1250) — compile-verified
//
#include <hip/hip_runtime.h>

typedef __attribute__((ext_vector_type(2))) float v2f;
typedef __attribute__((ext_vector_type(8))) float v8f;

#define BASE_VOCAB 32000
#define SVG_VOCAB  40000
#define TOTAL_VOCAB 72000
#define EMBED_DIM  2048
#define NTOK       2048            // B*S = 2*1024
#define TILE_K     256
#define LDS_STRIDE 258             // 256 + 2 pad: avoids 64-bank conflicts

// ---------------------------------------------------------------------------
// Kernel 1: embedding gather with SVG swap. One block per token row.
// hidden[t][d] = is_svg ? svg_embed[id-32000][d] : base_embed[id][d]
// ---------------------------------------------------------------------------
__global__ __launch_bounds__(256) void gather_embed(
    const int* __restrict__ input_ids, const int* __restrict__ svg_idx,
    const float* __restrict__ base_embed, const float* __restrict__ svg_embed,
    float* __restrict__ hidden, int* __restrict__ flags)
{
  const int t = blockIdx.x;
  const int id = input_ids[t];
  const int is_svg = (svg_idx[t] == 1) && (id >= BASE_VOCAB) && (id < TOTAL_VOCAB);
  int sid = id - BASE_VOCAB;
  sid = sid < 0 ? 0 : (sid >= SVG_VOCAB ? SVG_VOCAB - 1 : sid);
  const float* src = is_svg ? (svg_embed + (size_t)sid * EMBED_DIM)
                            : (base_embed + (size_t)id * EMBED_DIM);
  float* dst = hidden + (size_t)t * EMBED_DIM;
  // 2048 floats = 512 float4, 256 threads x 2
  const float4* s4 = (const float4*)src;
  float4* d4 = (float4*)dst;
  d4[threadIdx.x]       = s4[threadIdx.x];
  d4[threadIdx.x + 256] = s4[threadIdx.x + 256];
  if (threadIdx.x == 0) flags[t] = is_svg;
}

// ---------------------------------------------------------------------------
// Kernel 2: fp32 WMMA GEMM  out[m, outColOff + n]  =  select(hidden @ W)
//   hidden: [NTOK, 2048] row-major (L2-resident, 16 MB)
//   W     : [2048, N]    row-major (panel streamed ~once thanks to grid order)
// Block = 8 waves; block tile = 16 (M) x 512 (N); wave tile = 16 x 64.
// blockIdx.x = M tile (fast-varying -> blocks sharing the same 4 MB W panel
// co-schedule, keeping the panel hot in the 192 MB L2).
// mode 0: full store; svg rows with col<32000 get -1e30.
// mode 1: store only at svg rows (svg_proj slice), cols shifted by outColOff.
// ---------------------------------------------------------------------------
__global__ __launch_bounds__(256) void gemm_wmma(
    const float* __restrict__ hidden, const int* __restrict__ flags,
    const float* __restrict__ W, int N,
    float* __restrict__ out, int outColOff, int mode)
{
  __shared__ float As[16 * LDS_STRIDE];

  const int m0   = blockIdx.x * 16;
  const int wave = threadIdx.x >> 5;
  const int lane = threadIdx.x & 31;
  const int ln   = lane & 15;
  const int hi   = lane >> 4;          // 0: K even pair / M 0..7, 1: K odd pair / M 8..15
  const int n0   = (blockIdx.y * 8 + wave) * 64;
  const bool active = (n0 < N);        // wave-uniform

  // Per-lane 32-bit element offset into a W row-pair; wave-uniform row base is
  // advanced per k so the backend can use saddr + 32-bit voffset addressing.
  const int boff = 2 * hi * N + n0 + ln;

  v8f c0 = {}, c1 = {}, c2 = {}, c3 = {};

  for (int kc = 0; kc < EMBED_DIM; kc += TILE_K) {
    // cooperative stage of A tile: 16 rows x 256 K  (1024 float4 / 256 threads)
    #pragma unroll
    for (int s = threadIdx.x; s < 16 * (TILE_K / 4); s += 256) {
      const int row = s >> 6;          // 64 float4 per row
      const int c4  = s & 63;
      const float4 v = *(const float4*)(hidden + (size_t)(m0 + row) * EMBED_DIM + kc + c4 * 4);
      float* d = &As[row * LDS_STRIDE + c4 * 4];
      d[0] = v.x; d[1] = v.y; d[2] = v.z; d[3] = v.w;
    }
    __syncthreads();

    if (active) {
      #pragma unroll 4
      for (int kk = 0; kk < TILE_K; kk += 4) {
        // A 16x4: lanes 0-15 rows M=0..15 K={kk,kk+1}; lanes 16-31 K={kk+2,kk+3}
        const v2f a = *(const v2f*)&As[ln * LDS_STRIDE + kk + 2 * hi];
        // B 4x16 per 16-col tile: uniform base + lane offset (saddr form)
        const float* wrow = W + (size_t)(kc + kk) * (size_t)N;   // wave-uniform
        v2f b0; b0.x = wrow[boff];      b0.y = wrow[boff + N];
        v2f b1; b1.x = wrow[boff + 16]; b1.y = wrow[boff + N + 16];
        v2f b2; b2.x = wrow[boff + 32]; b2.y = wrow[boff + N + 32];
        v2f b3; b3.x = wrow[boff + 48]; b3.y = wrow[boff + N + 48];
        c0 = __builtin_amdgcn_wmma_f32_16x16x4_f32(false, a, false, b0, (short)0, c0, false, false);
        c1 = __builtin_amdgcn_wmma_f32_16x16x4_f32(false, a, false, b1, (short)0, c1, false, false);
        c2 = __builtin_amdgcn_wmma_f32_16x16x4_f32(false, a, false, b2, (short)0, c2, false, false);
        c3 = __builtin_amdgcn_wmma_f32_16x16x4_f32(false, a, false, b3, (short)0, c3, false, false);
      }
    }
    __syncthreads();
  }

  if (!active) return;

  // epilogue: C/D layout -> element i of acc maps to row m0 + i + 8*hi, col base + ln
  const int* frow = flags + m0;              // uniform base, lane offset below
  int f[8];
  #pragma unroll
  for (int i = 0; i < 8; ++i) f[i] = frow[i + 8 * hi];

  // uniform output base for this M tile; per-lane 32-bit element offset
  float* orow = out + (size_t)m0 * TOTAL_VOCAB + outColOff;
  #pragma unroll
  for (int t = 0; t < 4; ++t) {
    const v8f acc = (t == 0) ? c0 : (t == 1) ? c1 : (t == 2) ? c2 : c3;
    const int colIn = n0 + t * 16 + ln;      // column within this GEMM
    #pragma unroll
    for (int i = 0; i < 8; ++i) {
      const int off = (i + 8 * hi) * TOTAL_VOCAB + colIn;   // < 1.2M elements
      float v = acc[i];
      if (mode == 0) {
        if (f[i] && colIn < BASE_VOCAB) v = -1e30f;  // NEG_INF mask at svg rows
        __builtin_nontemporal_store(v, orow + off); // write-once: keep L2 for W/hidden
      } else {
        if (f[i]) __builtin_nontemporal_store(v, orow + off);
      }
    }
  }
}

// ---------------------------------------------------------------------------
extern "C" void kernel_launch(void* const* d_in, const int* in_sizes, int n_in,
                              void* d_out, int out_size, void* d_ws, size_t ws_size,
                              hipStream_t stream) {
  const int*   input_ids = (const int*)d_in[0];
  const int*   svg_idx   = (const int*)d_in[1];
  const float* base_emb  = (const float*)d_in[2];
  const float* svg_emb   = (const float*)d_in[3];
  const float* lm_head   = (const float*)d_in[4];
  const float* svg_proj  = (const float*)d_in[5];
  float* out = (float*)d_out;

  float* hidden = (float*)d_ws;                                   // 16 MB
  int*   flags  = (int*)((char*)d_ws + (size_t)NTOK * EMBED_DIM * sizeof(float));

  gather_embed<<<NTOK, 256, 0, stream>>>(input_ids, svg_idx, base_emb, svg_emb,
                                         hidden, flags);

  // Pass 1: full lm_head logits + -inf mask.  grid.x = M tiles (panel reuse).
  dim3 g1(NTOK / 16, (TOTAL_VOCAB + 511) / 512);   // (128, 141)
  gemm_wmma<<<g1, 256, 0, stream>>>(hidden, flags, lm_head, TOTAL_VOCAB, out, 0, 0);

  // Pass 2: svg_proj logits scattered into cols [32000, 72000) at svg rows.
  dim3 g2(NTOK / 16, (SVG_VOCAB + 511) / 512);     // (128, 79)
  gemm_wmma<<<g2, 256, 0, stream>>>(hidden, flags, svg_proj, SVG_VOCAB, out, BASE_VOCAB, 1);
}